// MultiHeadAttention_9775345566313
// MI455X (gfx1250) — compile-verified
//
#include <hip/hip_runtime.h>
#include <math.h>

typedef __attribute__((ext_vector_type(2))) float v2f;
typedef __attribute__((ext_vector_type(8))) float v8f;

#define BQ   4
#define SEQ  2048
#define EMB  768
#define NH   12
#define HD   64
#define NQKV 2304

__device__ __forceinline__ v8f wmma_f32(v2f a, v2f b, v8f c) {
  // D = A(16x4,f32) * B(4x16,f32) + C(16x16,f32), full fp32 precision
  return __builtin_amdgcn_wmma_f32_16x16x4_f32(false, a, false, b, (short)0, c,
                                               false, false);
}

// ---------------------------------------------------------------------------
// Kernel 1: qkv = X @ W_qkv, scattered into Q/K/V in [b][h][s][d] layout.
// Q pre-scaled by 1/sqrt(64). Wave computes a 64x64 tile (4x4 WMMA accs).
// Since n0 is 64-aligned, the whole tile maps to one (head, q/k/v) block.
// ---------------------------------------------------------------------------
__global__ void __launch_bounds__(128)
qkv_gemm_kernel(const float* __restrict__ X, const float* __restrict__ W,
                float* __restrict__ Qp, float* __restrict__ Kp,
                float* __restrict__ Vp) {
  const int lane = threadIdx.x & 31;
  const int wave = threadIdx.x >> 5;
  const int lq   = lane & 15;
  const int half = lane >> 4;
  const int m0 = blockIdx.x * 64;
  const int n0 = blockIdx.y * 256 + wave * 64;

  v8f acc[4][4] = {};
  const float* xr[4];
#pragma unroll
  for (int mi = 0; mi < 4; ++mi)
    xr[mi] = X + (size_t)(m0 + 16 * mi + lq) * EMB;

#pragma unroll 2
  for (int k0 = 0; k0 < EMB; k0 += 4) {
    const int kk = k0 + 2 * half;
    v2f a[4];
#pragma unroll
    for (int mi = 0; mi < 4; ++mi) a[mi] = *(const v2f*)(xr[mi] + kk);
    const float* wp = W + (size_t)kk * NQKV;
    v2f bb[4];
#pragma unroll
    for (int ni = 0; ni < 4; ++ni) {
      bb[ni].x = wp[n0 + 16 * ni + lq];
      bb[ni].y = wp[NQKV + n0 + 16 * ni + lq];
    }
#pragma unroll
    for (int mi = 0; mi < 4; ++mi)
#pragma unroll
      for (int ni = 0; ni < 4; ++ni)
        acc[mi][ni] = wmma_f32(a[mi], bb[ni], acc[mi][ni]);
  }

  // whole 64-wide tile lies in one (h, kind) block since n0 % 64 == 0
  const int h    = n0 / 192;
  const int rr   = n0 % 192;
  const int kind = rr >> 6;                  // 0=q 1=k 2=v
  float* dst = (kind == 0) ? Qp : ((kind == 1) ? Kp : Vp);
  const float scale = (kind == 0) ? 0.125f : 1.0f;   // 1/sqrt(64) into Q
#pragma unroll
  for (int mi = 0; mi < 4; ++mi) {
#pragma unroll
    for (int v = 0; v < 8; ++v) {
      const int r = m0 + 16 * mi + v + 8 * half;
      const int b = r >> 11;                 // / SEQ
      const int s = r & (SEQ - 1);
      float* drow = dst + (((size_t)b * NH + h) * SEQ + s) * HD + lq;
#pragma unroll
      for (int ni = 0; ni < 4; ++ni)
        drow[16 * ni] = acc[mi][ni][v] * scale;
    }
  }
}

// ---------------------------------------------------------------------------
// Kernel 2: flash attention. One wave per 32 query rows of one (b,h).
// K/V fragments shared across both 16-row blocks. Row sums kept as per-lane
// partials (corr is uniform per half-group) and reduced once at the end.
// ---------------------------------------------------------------------------
__global__ void __launch_bounds__(128)
attn_kernel(const float* __restrict__ Qp, const float* __restrict__ Kp,
            const float* __restrict__ Vp, const int* __restrict__ mask,
            float* __restrict__ AO) {
  __shared__ float ldsP[4 * 512];            // 32x16 fp32 P tile per wave
  const int lane = threadIdx.x & 31;
  const int wave = threadIdx.x >> 5;
  const int lq   = lane & 15;
  const int half = lane >> 4;

  const int bh     = blockIdx.x >> 4;        // 16 q-chunks of 128 per (b,h)
  const int qchunk = blockIdx.x & 15;
  const int q0     = qchunk * 128 + wave * 32;
  const int b      = bh / NH;

  const float* qptr = Qp + (size_t)bh * SEQ * HD;
  const float* kptr = Kp + (size_t)bh * SEQ * HD;
  const float* vptr = Vp + (size_t)bh * SEQ * HD;
  const int*   mrow = mask + b * SEQ;
  float* lds = ldsP + wave * 512;

  // Q strip 32x64 as 2x16 A-fragments (K-steps of 4)
  v2f qa[2][16];
#pragma unroll
  for (int rb = 0; rb < 2; ++rb)
#pragma unroll
    for (int kk = 0; kk < 16; ++kk)
      qa[rb][kk] =
          *(const v2f*)(qptr + (size_t)(q0 + 16 * rb + lq) * HD + 4 * kk + 2 * half);

  v8f o[2][4] = {};
  float mst[2][8], lpart[2][8];
#pragma unroll
  for (int rb = 0; rb < 2; ++rb)
#pragma unroll
    for (int v = 0; v < 8; ++v) { mst[rb][v] = -1e30f; lpart[rb][v] = 0.0f; }

  for (int kt = 0; kt < SEQ / 16; ++kt) {
    const int ky = kt * 16;

    // S(32x16) = Q(32x64) * K^T(64x16); K-fragment shared by both row blocks
    v8f sc[2] = {};
    const float* kb = kptr + (size_t)(ky + lq) * HD + 2 * half;
#pragma unroll
    for (int kk = 0; kk < 16; ++kk) {
      v2f bf = *(const v2f*)(kb + 4 * kk);
      sc[0] = wmma_f32(qa[0][kk], bf, sc[0]);
      sc[1] = wmma_f32(qa[1][kk], bf, sc[1]);
    }

    const int mk = mrow[ky + lq];            // this lane's key column mask
#pragma unroll
    for (int rb = 0; rb < 2; ++rb) {
#pragma unroll
      for (int v = 0; v < 8; ++v) {
        float s = mk ? sc[rb][v] : -1e30f;
        // row max across the 16 lanes of this half-group
        float mx = s;
        mx = fmaxf(mx, __shfl_xor(mx, 1, 32));
        mx = fmaxf(mx, __shfl_xor(mx, 2, 32));
        mx = fmaxf(mx, __shfl_xor(mx, 4, 32));
        mx = fmaxf(mx, __shfl_xor(mx, 8, 32));
        const float mnew = fmaxf(mst[rb][v], mx);
        const float corr = __expf(mst[rb][v] - mnew);   // uniform in half-group
        const float pe   = __expf(s - mnew);
        lpart[rb][v] = lpart[rb][v] * corr + pe;        // per-lane partial sum
        mst[rb][v]   = mnew;
        o[rb][0][v] *= corr; o[rb][1][v] *= corr;
        o[rb][2][v] *= corr; o[rb][3][v] *= corr;
        // P tile: C/D layout -> LDS row-major (per-wave region, in-order LDS)
        lds[(rb * 16 + v + 8 * half) * 16 + lq] = pe;
      }
    }
    __asm__ volatile("" ::: "memory");

    // O(32x64) += P(32x16) * V(16x64); V-fragment shared by both row blocks
#pragma unroll
    for (int kk2 = 0; kk2 < 4; ++kk2) {
      v2f pa0 = *(const v2f*)(lds + (lq)      * 16 + 4 * kk2 + 2 * half);
      v2f pa1 = *(const v2f*)(lds + (16 + lq) * 16 + 4 * kk2 + 2 * half);
      const float* vb = vptr + (size_t)(ky + 4 * kk2 + 2 * half) * HD + lq;
#pragma unroll
      for (int nt = 0; nt < 4; ++nt) {
        v2f bv;
        bv.x = vb[16 * nt];
        bv.y = vb[HD + 16 * nt];
        o[0][nt] = wmma_f32(pa0, bv, o[0][nt]);
        o[1][nt] = wmma_f32(pa1, bv, o[1][nt]);
      }
    }
  }

  // final cross-lane reduction of row sums, normalize, write (B,S,E) layout
  const int hcol = (bh % NH) * HD;
#pragma unroll
  for (int rb = 0; rb < 2; ++rb) {
#pragma unroll
    for (int v = 0; v < 8; ++v) {
      float ps = lpart[rb][v];
      ps += __shfl_xor(ps, 1, 32);
      ps += __shfl_xor(ps, 2, 32);
      ps += __shfl_xor(ps, 4, 32);
      ps += __shfl_xor(ps, 8, 32);
      const float inv = 1.0f / ps;
      const int srow  = q0 + 16 * rb + v + 8 * half;
      float* orow = AO + ((size_t)b * SEQ + srow) * EMB + hcol + lq;
      orow[0]  = o[rb][0][v] * inv;
      orow[16] = o[rb][1][v] * inv;
      orow[32] = o[rb][2][v] * inv;
      orow[48] = o[rb][3][v] * inv;
    }
  }
}

// ---------------------------------------------------------------------------
// Kernel 3: out = AO @ W_fc. Wave computes a 64x64 tile.
// ---------------------------------------------------------------------------
__global__ void __launch_bounds__(128)
fc_gemm_kernel(const float* __restrict__ A, const float* __restrict__ W,
               float* __restrict__ C) {
  const int lane = threadIdx.x & 31;
  const int wave = threadIdx.x >> 5;
  const int lq   = lane & 15;
  const int half = lane >> 4;
  const int m0 = blockIdx.x * 64;
  const int n0 = blockIdx.y * 256 + wave * 64;

  v8f acc[4][4] = {};
  const float* ar[4];
#pragma unroll
  for (int mi = 0; mi < 4; ++mi)
    ar[mi] = A + (size_t)(m0 + 16 * mi + lq) * EMB;

#pragma unroll 2
  for (int k0 = 0; k0 < EMB; k0 += 4) {
    const int kk = k0 + 2 * half;
    v2f a[4];
#pragma unroll
    for (int mi = 0; mi < 4; ++mi) a[mi] = *(const v2f*)(ar[mi] + kk);
    const float* wp = W + (size_t)kk * EMB;
    v2f bb[4];
#pragma unroll
    for (int ni = 0; ni < 4; ++ni) {
      bb[ni].x = wp[n0 + 16 * ni + lq];
      bb[ni].y = wp[EMB + n0 + 16 * ni + lq];
    }
#pragma unroll
    for (int mi = 0; mi < 4; ++mi)
#pragma unroll
      for (int ni = 0; ni < 4; ++ni)
        acc[mi][ni] = wmma_f32(a[mi], bb[ni], acc[mi][ni]);
  }

#pragma unroll
  for (int mi = 0; mi < 4; ++mi) {
#pragma unroll
    for (int v = 0; v < 8; ++v) {
      const int r = m0 + 16 * mi + v + 8 * half;
      float* crow = C + (size_t)r * EMB + n0 + lq;
#pragma unroll
      for (int ni = 0; ni < 4; ++ni)
        crow[16 * ni] = acc[mi][ni][v];
    }
  }
}

// ---------------------------------------------------------------------------
extern "C" void kernel_launch(void* const* d_in, const int* in_sizes, int n_in,
                              void* d_out, int out_size, void* d_ws, size_t ws_size,
                              hipStream_t stream) {
  const float* X     = (const float*)d_in[0];
  const int*   mask  = (const int*)d_in[1];
  const float* W_qkv = (const float*)d_in[2];
  const float* W_fc  = (const float*)d_in[3];
  float* out = (float*)d_out;

  const size_t per = (size_t)BQ * NH * SEQ * HD;   // elements per Q/K/V tensor
  float* Qp = (float*)d_ws;
  float* Kp = Qp + per;
  float* Vp = Kp + per;
  float* AO = Vp + per;

  // qkv projection: M=8192 x N=2304, 64x64 tile per wave
  qkv_gemm_kernel<<<dim3(8192 / 64, NQKV / 256), 128, 0, stream>>>(X, W_qkv, Qp, Kp, Vp);
  // flash attention: B*H*(S/128) blocks, 4 waves each handling 32 query rows
  attn_kernel<<<dim3(BQ * NH * (SEQ / 128)), 128, 0, stream>>>(Qp, Kp, Vp, mask, AO);
  // final projection: M=8192 x N=768, 64x64 tile per wave
  fc_gemm_kernel<<<dim3(8192 / 64, EMB / 256), 128, 0, stream>>>(AO, W_fc, out);
}